// ROIAlign_42502996361807
// MI455X (gfx1250) — compile-verified
//
#include <hip/hip_runtime.h>

typedef __attribute__((ext_vector_type(2))) float v2f;
typedef __attribute__((ext_vector_type(8))) float v8f;

#ifndef __has_builtin
#define __has_builtin(x) 0
#endif

#if defined(__AMDGCN__) && __has_builtin(__builtin_amdgcn_wmma_f32_16x16x4_f32)
#define ROI_USE_WMMA 1
#else
#define ROI_USE_WMMA 0
#endif

namespace {
constexpr int P       = 12;
constexpr int PP      = P * P;            // 144 bins per ROI
constexpr int C       = 256;
constexpr int H       = 200;
constexpr int W       = 200;
constexpr int HW      = H * W;
constexpr int CSPLIT  = 2;                // channel halves per ROI
constexpr int CCHUNK  = C / CSPLIT;       // 128
constexpr int THREADS = 256;              // 8 wave32s
constexpr int ITERS   = CCHUNK * PP / THREADS;  // 72
constexpr float SCALE = 0.25f;            // 1 / SPATIAL_SCALE
}

__global__ __launch_bounds__(THREADS)
void roi_align_kernel(const float* __restrict__ x,
                      const float* __restrict__ rois,
                      float* __restrict__ out)
{
    __shared__ int   sRow0[P], sRow1[P], sX0[P], sX1[P];
    __shared__ float sW[4][PP];           // w00, w01, w10, w11 per bin
    __shared__ int   sB;

    const int n   = blockIdx.x;
    const int tid = threadIdx.x;
    const float* __restrict__ r = rois + n * 5;

#if ROI_USE_WMMA
    // ---- Phase 1 (wave 0): geometry + all 576 bilinear weights via four
    //      exact f32 rank-1 outer products on the matrix unit. ----
    if (tid < 32) {
        const float x1r = r[1] * SCALE, y1r = r[2] * SCALE;
        const float x2r = r[3] * SCALE, y2r = r[4] * SCALE;
        const float bh = (y2r - y1r) * (1.0f / (float)P);
        const float bw = (x2r - x1r) * (1.0f / (float)P);
        const float cc = (float)tid + 0.5f;        // lane = py index and px index
        const float ysv = fminf(fmaxf(y1r + cc * bh, 0.0f), (float)(H - 1));
        const float xsv = fminf(fmaxf(x1r + cc * bw, 0.0f), (float)(W - 1));
        const float y0f = floorf(ysv), x0f = floorf(xsv);
        const float ly = ysv - y0f,   lx = xsv - x0f;
        const int   y0 = (int)y0f,    x0 = (int)x0f;
        const int   y1i = min(y0 + 1, H - 1), x1i = min(x0 + 1, W - 1);
        if (tid < P) {
            sRow0[tid] = y0  * W;
            sRow1[tid] = y1i * W;
            sX0[tid]   = x0;
            sX1[tid]   = x1i;
        }
        if (tid == 0) sB = (int)r[0];

        // A (16x4 f32): lanes 0-15 hold A[M=lane][K=0]; everything else zero.
        // B (4x16 f32): lanes 0-15 of VGPR0 hold B[K=0][N=lane]; rest zero.
        const bool lo = (tid < 16);
        v2f ay0; ay0.x = lo ? (1.0f - ly) : 0.0f; ay0.y = 0.0f;
        v2f ay1; ay1.x = lo ? ly          : 0.0f; ay1.y = 0.0f;
        v2f bx0; bx0.x = lo ? (1.0f - lx) : 0.0f; bx0.y = 0.0f;
        v2f bx1; bx1.x = lo ? lx          : 0.0f; bx1.y = 0.0f;
        v8f z = {};
        // D[m][n] = wy[m] * wx[n]  (exact: single f32 product per element)
        v8f d00 = __builtin_amdgcn_wmma_f32_16x16x4_f32(false, ay0, false, bx0, (short)0, z, false, false);
        v8f d01 = __builtin_amdgcn_wmma_f32_16x16x4_f32(false, ay0, false, bx1, (short)0, z, false, false);
        v8f d10 = __builtin_amdgcn_wmma_f32_16x16x4_f32(false, ay1, false, bx0, (short)0, z, false, false);
        v8f d11 = __builtin_amdgcn_wmma_f32_16x16x4_f32(false, ay1, false, bx1, (short)0, z, false, false);

        // C/D 16x16 f32 layout: lanes 0-15 -> N=lane, VGPR k -> M=k;
        //                       lanes 16-31 -> N=lane-16, VGPR k -> M=8+k.
        const int col    = tid & 15;
        const int rowOff = lo ? 0 : 8;
        if (col < P) {
#pragma unroll
            for (int rr = 0; rr < 8; ++rr) {
                const int row = rowOff + rr;
                if (row < P) {
                    sW[0][row * P + col] = d00[rr];
                    sW[1][row * P + col] = d01[rr];
                    sW[2][row * P + col] = d10[rr];
                    sW[3][row * P + col] = d11[rr];
                }
            }
        }
    }
#else
    // ---- Phase 1 fallback: VALU weight generation ----
    {
        const float x1r = r[1] * SCALE, y1r = r[2] * SCALE;
        const float x2r = r[3] * SCALE, y2r = r[4] * SCALE;
        const float bh = (y2r - y1r) * (1.0f / (float)P);
        const float bw = (x2r - x1r) * (1.0f / (float)P);
        if (tid < PP) {
            const int py = tid / P, px = tid - py * P;
            const float ysv = fminf(fmaxf(y1r + ((float)py + 0.5f) * bh, 0.0f), (float)(H - 1));
            const float xsv = fminf(fmaxf(x1r + ((float)px + 0.5f) * bw, 0.0f), (float)(W - 1));
            const float ly = ysv - floorf(ysv);
            const float lx = xsv - floorf(xsv);
            sW[0][tid] = (1.0f - ly) * (1.0f - lx);
            sW[1][tid] = (1.0f - ly) * lx;
            sW[2][tid] = ly * (1.0f - lx);
            sW[3][tid] = ly * lx;
        }
        if (tid < P) {
            const float cc = (float)tid + 0.5f;
            const float ysv = fminf(fmaxf(y1r + cc * bh, 0.0f), (float)(H - 1));
            const float xsv = fminf(fmaxf(x1r + cc * bw, 0.0f), (float)(W - 1));
            const int y0 = (int)floorf(ysv);
            const int x0 = (int)floorf(xsv);
            sRow0[tid] = y0 * W;
            sRow1[tid] = min(y0 + 1, H - 1) * W;
            sX0[tid]   = x0;
            sX1[tid]   = min(x0 + 1, W - 1);
        }
        if (tid == 0) sB = (int)r[0];
    }
#endif
    __syncthreads();

    // ---- Phase 2: gather + weighted sum. px varies fastest across threads ->
    //      coalesced NT stores; corner loads cluster in <=4 py rows per wave. ----
    const int cbase = blockIdx.y * CCHUNK;
    const float* __restrict__ xb = x + ((size_t)sB * C + cbase) * (size_t)HW;
    float* __restrict__ ob = out + ((size_t)n * C + cbase) * (size_t)PP;

#pragma unroll 2
    for (int k = 0; k < ITERS; ++k) {
        const int j   = k * THREADS + tid;
        const int c   = j / PP;
        const int bin = j - c * PP;
        const int py  = bin / P;
        const int px  = bin - py * P;

        const int r0 = sRow0[py], r1 = sRow1[py];
        const int c0 = sX0[px],   c1 = sX1[px];
        const float* __restrict__ p = xb + (size_t)c * HW;

        const float v00 = p[r0 + c0];
        const float v01 = p[r0 + c1];
        const float v10 = p[r1 + c0];
        const float v11 = p[r1 + c1];

        const float acc = sW[0][bin] * v00 + sW[1][bin] * v01
                        + sW[2][bin] * v10 + sW[3][bin] * v11;

        // Write-once output: non-temporal so x stays resident in the 192MB L2.
        __builtin_nontemporal_store(acc, ob + j);
    }
}

extern "C" void kernel_launch(void* const* d_in, const int* in_sizes, int n_in,
                              void* d_out, int out_size, void* d_ws, size_t ws_size,
                              hipStream_t stream) {
    (void)n_in; (void)out_size; (void)d_ws; (void)ws_size;
    const float* x    = (const float*)d_in[0];
    const float* rois = (const float*)d_in[1];
    float* out        = (float*)d_out;
    const int N = in_sizes[1] / 5;   // rois is (N, 5)
    dim3 grid((unsigned)N, CSPLIT, 1);
    roi_align_kernel<<<grid, THREADS, 0, stream>>>(x, rois, out);
}